// weightedBCE_75582834475452
// MI455X (gfx1250) — compile-verified
//
#include <hip/hip_runtime.h>
#include <hip/hip_bf16.h>

typedef float v4f __attribute__((ext_vector_type(4)));
typedef int   v4i __attribute__((ext_vector_type(4)));
typedef float v2f __attribute__((ext_vector_type(2)));
typedef float v8f __attribute__((ext_vector_type(8)));

#define NB_BLOCKS 4096   // must be a multiple of 64 for the WMMA reduce
#define NT_THREADS 256

// ---------------------------------------------------------------------------
// Pass 1: streaming weighted-BCE partial sums, one float per block into d_ws.
// Bandwidth-bound: 16B x + 16B y per lane-iter via b128 nontemporal loads,
// one v_log_f32 per element, branchless selects, wave32 shuffle reduction.
// ---------------------------------------------------------------------------
__global__ void wbce_partial_kernel(const float* __restrict__ x,
                                    const int* __restrict__ y,
                                    const float* __restrict__ w,
                                    float* __restrict__ partials,
                                    int n) {
    const int tid    = blockIdx.x * blockDim.x + threadIdx.x;
    const int stride = gridDim.x * blockDim.x;

    const float w0 = w[0];
    const float w1 = w[1];

    float acc_pos = 0.0f;   // sum of log(x)   where y==1
    float acc_neg = 0.0f;   // sum of log(1-x) where y==0

    const v4f* __restrict__ x4 = (const v4f*)x;
    const v4i* __restrict__ y4 = (const v4i*)y;
    const int n4 = n >> 2;

    for (int i = tid; i < n4; i += stride) {
        v4f xv = __builtin_nontemporal_load(x4 + i);
        v4i yv = __builtin_nontemporal_load(y4 + i);
#pragma unroll
        for (int c = 0; c < 4; ++c) {
            const float xc  = xv[c];
            const bool  pos = (yv[c] == 1);
            const float t   = pos ? xc : (1.0f - xc);
            const float l   = __logf(t);          // v_log_f32 + mul
            acc_pos += pos ? l : 0.0f;
            acc_neg += pos ? 0.0f : l;
        }
    }
    // scalar tail (dead for N % 4 == 0, kept for generality)
    for (int i = (n4 << 2) + tid; i < n; i += stride) {
        const float xc  = x[i];
        const bool  pos = (y[i] == 1);
        const float t   = pos ? xc : (1.0f - xc);
        const float l   = __logf(t);
        acc_pos += pos ? l : 0.0f;
        acc_neg += pos ? 0.0f : l;
    }

    // loss contribution of this thread (negate the log-sums, apply weights once)
    float part = -(w1 * acc_pos + w0 * acc_neg);

    // wave32 reduction
#pragma unroll
    for (int off = 16; off > 0; off >>= 1)
        part += __shfl_down(part, off, 32);

    __shared__ float lds[NT_THREADS / 32];
    const int lane = threadIdx.x & 31;
    const int wid  = threadIdx.x >> 5;
    if (lane == 0) lds[wid] = part;
    __syncthreads();

    if (wid == 0) {
        float v = (lane < (blockDim.x >> 5)) ? lds[lane] : 0.0f;
#pragma unroll
        for (int off = 4; off > 0; off >>= 1)
            v += __shfl_down(v, off, 32);
        if (lane == 0) partials[blockIdx.x] = v;
    }
}

// ---------------------------------------------------------------------------
// Pass 2: reduce NB partials with V_WMMA_F32_16X16X4_F32 (ones-matrix trick).
// One wave (32 lanes, EXEC all-ones as WMMA requires).
// A(16x4) tile = 64 consecutive partials, row-major: A[m,k] = p[m*4+k].
// ISA A-layout (32-bit 16x4): VGPR0 = {K=0 | lanes 0-15, K=2 | lanes 16-31},
// VGPR1 = {K=1, K=3}  ->  lane l loads float2 at (l&15)*4 + (l>>4)*2.
// With B = ones: D[m,n] = sum_k A[m,k] + C[m,n]; after the loop every column
// of C holds per-row totals, so grand total = s(lane) + s(lane^16) where s is
// the per-lane sum of the 8 C VGPRs.
// ---------------------------------------------------------------------------
__global__ void wbce_reduce_wmma_kernel(const float* __restrict__ partials,
                                        float* __restrict__ out,
                                        int nb, float inv_n) {
    const int lane = threadIdx.x;              // 0..31, one full wave
    const int base = (lane & 15) * 4 + (lane >> 4) * 2;

    v2f ones; ones.x = 1.0f; ones.y = 1.0f;    // B matrix: all ones
    v8f c = {};

    for (int t = 0; t < nb; t += 64) {
        const float* p = partials + t + base;
        v2f a; a.x = p[0]; a.y = p[1];
        // 8 args: (neg_a, A, neg_b, B, c_mod, C, reuse_a, reuse_b)
        c = __builtin_amdgcn_wmma_f32_16x16x4_f32(
                /*neg_a=*/false, a, /*neg_b=*/false, ones,
                /*c_mod=*/(short)0, c, /*reuse_a=*/false, /*reuse_b=*/false);
    }

    float s = c[0] + c[1] + c[2] + c[3] + c[4] + c[5] + c[6] + c[7];
    s += __shfl_xor(s, 16, 32);                // full column sum == grand total

    if (lane == 0) out[0] = s * inv_n;
}

extern "C" void kernel_launch(void* const* d_in, const int* in_sizes, int n_in,
                              void* d_out, int out_size, void* d_ws, size_t ws_size,
                              hipStream_t stream) {
    const float* x = (const float*)d_in[0];
    const int*   y = (const int*)d_in[1];
    const float* w = (const float*)d_in[2];
    float* out      = (float*)d_out;
    float* partials = (float*)d_ws;            // NB_BLOCKS floats (16 KB)
    const int n = in_sizes[0];

    wbce_partial_kernel<<<NB_BLOCKS, NT_THREADS, 0, stream>>>(x, y, w, partials, n);
    wbce_reduce_wmma_kernel<<<1, 32, 0, stream>>>(partials, out, NB_BLOCKS,
                                                  1.0f / (float)n);
}